// SequenceTagger_33878702030954
// MI455X (gfx1250) — compile-verified
//
#include <hip/hip_runtime.h>

typedef __attribute__((ext_vector_type(2))) float v2f;
typedef __attribute__((ext_vector_type(8))) float v8f;

#define NINF (-100000.0f)

constexpr int B  = 64;
constexpr int T  = 512;
constexpr int H  = 768;
constexpr int N  = 25;
constexpr int NP = 32;                    // padded tag dim for scratch layout
constexpr int BT = B * T;                 // 32768 rows of the logits matrix
constexpr long LP_TOTAL = (long)B * T * N * N;   // 20,480,000 lp elements

// ---------------------------------------------------------------------------
// Kernel 0: pad W (H x 25) into Wp (H x 32), zero-filled columns 25..31.
// Removes all bounds checks / EXEC divergence from the GEMM inner loop.
// ---------------------------------------------------------------------------
__global__ __launch_bounds__(256) void crf_padW(
    const float* __restrict__ Wm,   // [H][N]
    float* __restrict__ Wp)         // [H][NP]
{
    int idx = blockIdx.x * blockDim.x + threadIdx.x;   // H*NP = 24576
    if (idx >= H * NP) return;
    const int h = idx >> 5;         // / NP
    const int c = idx & (NP - 1);   // % NP
    Wp[idx] = (c < N) ? Wm[h * N + c] : 0.0f;
}

// ---------------------------------------------------------------------------
// Kernel 1: logits = emb @ W + b via V_WMMA_F32_16X16X4_F32 (fp32, K-step 4).
// Each wave computes a 64x32 tile: 4 row tiles x 2 col tiles = 8 accumulators,
// so the 4 B-fragment loads per k-step feed 8 WMMAs. Embedding loads are
// non-temporal (streamed once); W/logits stay cache-resident.
// A layout (32b 16x4): lanes 0-15 K={k,k+1}; lanes 16-31 K={k+2,k+3}, M=lane&15.
// B layout (32b 4x16): v0/v1 = K rows, N=lane&15, lane halves split K like A.
// C layout: VGPR r -> M=r (lanes 0-15) / M=r+8 (lanes 16-31), N=lane&15.
// ---------------------------------------------------------------------------
__global__ __launch_bounds__(128) void crf_logits_wmma(
    const float* __restrict__ emb,     // [BT][H]
    const float* __restrict__ Wp,      // [H][NP] padded, zero cols >= N
    const float* __restrict__ bias,    // [N]
    float* __restrict__ logitsWS)      // [BT][NP]
{
    const int lane   = threadIdx.x & 31;
    const int wave   = threadIdx.x >> 5;
    const int rowGrp = blockIdx.x * 4 + wave;       // 512 groups of 64 rows
    const int kHalf  = (lane >> 4) << 1;            // 0 or 2
    const int col0   = lane & 15;                   // 0..15
    const int col1   = col0 + 16;                   // 16..31 (zero-padded in Wp)

    const float* A0 = emb + (long)(rowGrp * 64 +  0 + col0) * H;
    const float* A1 = emb + (long)(rowGrp * 64 + 16 + col0) * H;
    const float* A2 = emb + (long)(rowGrp * 64 + 32 + col0) * H;
    const float* A3 = emb + (long)(rowGrp * 64 + 48 + col0) * H;

    v8f acc00 = {}, acc01 = {};
    v8f acc10 = {}, acc11 = {};
    v8f acc20 = {}, acc21 = {};
    v8f acc30 = {}, acc31 = {};

#pragma unroll 2
    for (int k = 0; k < H; k += 4) {
        const int kr = k + kHalf;

        // B fragments (shared by all 4 row tiles), unconditional loads
        v2f b0; b0.x = Wp[kr * NP + col0]; b0.y = Wp[(kr + 1) * NP + col0];
        v2f b1; b1.x = Wp[kr * NP + col1]; b1.y = Wp[(kr + 1) * NP + col1];

        // A fragments: contiguous pair -> b64, non-temporal (streamed once)
        const v2f a0 = __builtin_nontemporal_load((const v2f*)(A0 + kr));
        const v2f a1 = __builtin_nontemporal_load((const v2f*)(A1 + kr));
        const v2f a2 = __builtin_nontemporal_load((const v2f*)(A2 + kr));
        const v2f a3 = __builtin_nontemporal_load((const v2f*)(A3 + kr));

        acc00 = __builtin_amdgcn_wmma_f32_16x16x4_f32(false, a0, false, b0, (short)0, acc00, false, false);
        acc01 = __builtin_amdgcn_wmma_f32_16x16x4_f32(false, a0, false, b1, (short)0, acc01, false, false);
        acc10 = __builtin_amdgcn_wmma_f32_16x16x4_f32(false, a1, false, b0, (short)0, acc10, false, false);
        acc11 = __builtin_amdgcn_wmma_f32_16x16x4_f32(false, a1, false, b1, (short)0, acc11, false, false);
        acc20 = __builtin_amdgcn_wmma_f32_16x16x4_f32(false, a2, false, b0, (short)0, acc20, false, false);
        acc21 = __builtin_amdgcn_wmma_f32_16x16x4_f32(false, a2, false, b1, (short)0, acc21, false, false);
        acc30 = __builtin_amdgcn_wmma_f32_16x16x4_f32(false, a3, false, b0, (short)0, acc30, false, false);
        acc31 = __builtin_amdgcn_wmma_f32_16x16x4_f32(false, a3, false, b1, (short)0, acc31, false, false);
    }

    const float bias0 = bias[col0];
    const float bias1 = (col1 < N) ? bias[col1] : 0.0f;
    const int   half  = (lane >> 4) << 3;   // +8 rows for upper lane half

    const v8f* accs[4][2] = { {&acc00, &acc01}, {&acc10, &acc11},
                              {&acc20, &acc21}, {&acc30, &acc31} };
#pragma unroll
    for (int tile = 0; tile < 4; ++tile) {
        const int mBase = rowGrp * 64 + tile * 16 + half;
#pragma unroll
        for (int r = 0; r < 8; ++r) {
            float* dst = logitsWS + (long)(mBase + r) * NP;
            dst[col0] = (*accs[tile][0])[r] + bias0;
            dst[col1] = (*accs[tile][1])[r] + bias1;
        }
    }
}

// ---------------------------------------------------------------------------
// Kernel 2: sequence lengths from the boolean mask (mask is a prefix mask).
// ---------------------------------------------------------------------------
__global__ __launch_bounds__(512) void crf_lengths(
    const unsigned char* __restrict__ mask,   // [B][T] (bool)
    int* __restrict__ lens)                   // [B]
{
    __shared__ int s;
    if (threadIdx.x == 0) s = 0;
    __syncthreads();
    if (mask[(long)blockIdx.x * T + threadIdx.x]) atomicAdd(&s, 1);
    __syncthreads();
    if (threadIdx.x == 0) lens[blockIdx.x] = s;
}

// ---------------------------------------------------------------------------
// Kernel 3: lp[b,t,i,j] = logits[b,t,j] + trans[i,j]  (+ start/end fixups).
// 82 MB write-once stream -> non-temporal stores; logits/trans are L2-hot.
// ---------------------------------------------------------------------------
__global__ __launch_bounds__(256) void crf_lp(
    const float* __restrict__ logitsWS,   // [BT][NP]
    const float* __restrict__ trans,      // [N][N]
    const float* __restrict__ startT,     // [N]
    const float* __restrict__ endT,       // [N]
    const int*   __restrict__ lens,       // [B]
    float* __restrict__ lp)               // [B][T][N][N]
{
    long idx = (long)blockIdx.x * blockDim.x + threadIdx.x;
    if (idx >= LP_TOTAL) return;

    const int  j  = (int)(idx % N);
    long       r  = idx / N;
    const int  i  = (int)(r % N);
    r            /= N;
    const int  t  = (int)(r % T);
    const int  b  = (int)(r / T);

    const float lg = logitsWS[((long)b * T + t) * NP + j];
    float v = (t == 0) ? (startT[j] + lg) : (lg + trans[i * N + j]);
    if (t == lens[b] - 1) v += endT[j];
    __builtin_nontemporal_store(v, &lp[idx]);
}

// ---------------------------------------------------------------------------
// Kernel 4: Viterbi. One wave per batch (wave32: lane j owns state j, j<25).
// Transition column (trans + NINF-constraint) lives in per-lane registers;
// alpha broadcast via __shfl; backpointers in LDS so the serial backward
// walk stays on-chip.
// ---------------------------------------------------------------------------
__global__ __launch_bounds__(32) void crf_viterbi(
    const float* __restrict__ logitsWS,        // [BT][NP]
    const float* __restrict__ trans,           // [N][N]
    const float* __restrict__ startT,          // [N]
    const float* __restrict__ endT,            // [N]
    const unsigned char* __restrict__ scon,    // [N] bool
    const unsigned char* __restrict__ econ,    // [N] bool
    const unsigned char* __restrict__ tcon,    // [N][N] bool
    const int*   __restrict__ lens,            // [B]
    const int*   __restrict__ padIdx,          // [1]
    float* __restrict__ tagsOut)               // [B][T] (as float values)
{
    __shared__ unsigned char bp_s[T][32];      // 16 KB backpointers

    const int b   = blockIdx.x;
    const int j   = threadIdx.x;
    const bool act = (j < N);

    // Per-lane column of (transitions + transition-constraint) for state j.
    float colM[N];
#pragma unroll
    for (int i = 0; i < N; ++i)
        colM[i] = act ? (trans[i * N + j] + (tcon[i * N + j] ? 0.0f : NINF))
                      : NINF;

    const int len = lens[b];
    const float* lgBase = logitsWS + (long)b * T * NP;

    // alpha0 = start_transitions + logits[b,0] + start-constraint
    float alpha = act ? (lgBase[j] + startT[j] + (scon[j] ? 0.0f : NINF))
                      : (NINF * 4.0f);

    for (int t = 1; t < T; ++t) {
        const bool inSeq = (t < len);
        float lg = 0.0f;
        if (inSeq && act) {
            lg = lgBase[(long)t * NP + j];
            if (t == len - 1) lg += endT[j] + (econ[j] ? 0.0f : NINF);
        }
        float best = -3.4e38f;
        int   arg  = 0;
#pragma unroll
        for (int i = 0; i < N; ++i) {
            const float ai  = __shfl(alpha, i, 32);
            const float add = inSeq ? colM[i] : (i == j ? 0.0f : NINF);
            const float s   = ai + add;
            if (s > best) { best = s; arg = i; }   // strict > keeps first max
        }
        if (act) alpha = best + lg;
        bp_s[t][j] = (unsigned char)arg;
    }

    // argmax of final alpha (replicated in all lanes)
    float bestA = -3.4e38f;
    int   lastTag = 0;
#pragma unroll
    for (int i = 0; i < N; ++i) {
        const float ai = __shfl(alpha, i, 32);
        if (ai > bestA) { bestA = ai; lastTag = i; }
    }

    if (j == 0) {
        const float padv = (float)padIdx[0];
        float* outRow = tagsOut + (long)b * T;
        int cur = lastTag;
        outRow[T - 1] = ((T - 1) < len) ? (float)cur : padv;
        for (int t = T - 1; t >= 1; --t) {
            cur = bp_s[t][cur];
            outRow[t - 1] = ((t - 1) < len) ? (float)cur : padv;
        }
    }
}

// ---------------------------------------------------------------------------
extern "C" void kernel_launch(void* const* d_in, const int* in_sizes, int n_in,
                              void* d_out, int out_size, void* d_ws, size_t ws_size,
                              hipStream_t stream) {
    const float*         emb    = (const float*)d_in[0];
    const float*         Wm     = (const float*)d_in[1];
    const float*         bias   = (const float*)d_in[2];
    const float*         trans  = (const float*)d_in[3];
    const float*         startT = (const float*)d_in[4];
    const float*         endT   = (const float*)d_in[5];
    const unsigned char* mask   = (const unsigned char*)d_in[6];
    const unsigned char* scon   = (const unsigned char*)d_in[7];
    const unsigned char* econ   = (const unsigned char*)d_in[8];
    const unsigned char* tcon   = (const unsigned char*)d_in[9];
    const int*           padIdx = (const int*)d_in[10];

    float* out = (float*)d_out;

    // workspace layout:
    //   [0, 4MB)        logitsWS  [BT][NP] f32
    //   [4MB, +256B)    lens      [B] int
    //   [4MB+1KB, +96K) Wp        [H][NP] f32 (padded W)
    char*  wsb      = (char*)d_ws;
    float* logitsWS = (float*)wsb;
    int*   lens     = (int*)(wsb + (size_t)BT * NP * sizeof(float));
    float* Wp       = (float*)(wsb + (size_t)BT * NP * sizeof(float) + 1024);

    // 0) pad W to 32 columns (zero-filled)
    crf_padW<<<(H * NP + 255) / 256, 256, 0, stream>>>(Wm, Wp);

    // 1) lengths from mask
    crf_lengths<<<B, T, 0, stream>>>(mask, lens);

    // 2) logits GEMM (WMMA fp32): 128 blocks x 4 waves, 64x32 tile per wave
    crf_logits_wmma<<<BT / 256, 128, 0, stream>>>(emb, Wp, bias, logitsWS);

    // 3) lp expansion (first tuple output)
    crf_lp<<<(int)((LP_TOTAL + 255) / 256), 256, 0, stream>>>(
        logitsWS, trans, startT, endT, lens, out);

    // 4) Viterbi decode (second tuple output, written after lp region)
    crf_viterbi<<<B, 32, 0, stream>>>(
        logitsWS, trans, startT, endT, scon, econ, tcon, lens, padIdx,
        out + LP_TOTAL);
}